// DeformScaledDotAttnLayerLocal_22222160789706
// MI455X (gfx1250) — compile-verified
//
#include <hip/hip_runtime.h>
#include <math.h>

// ---------------------------------------------------------------------------
// DeformScaledDotAttnLayerLocal for MI455X (gfx1250, wave32, WMMA f32 path)
// ---------------------------------------------------------------------------
typedef __attribute__((ext_vector_type(2))) float v2f;
typedef __attribute__((ext_vector_type(8))) float v8f;
typedef __attribute__((ext_vector_type(4))) int v4i;

#define WMMA_F32(a, b, c) \
  __builtin_amdgcn_wmma_f32_16x16x4_f32(false, (a), false, (b), (short)0, (c), false, false)

constexpr int Cn  = 256;    // NF
constexpr int HW  = 4096;   // 64*64
constexpr int NPt = 8;      // points
constexpr int DPH = 32;     // dim per head

// ---- gfx1250 async global->LDS staging (ASYNCcnt path), guarded ------------
#if __has_builtin(__builtin_amdgcn_global_load_async_to_lds_b128)
#define ASYNC_LDS 1
#else
#define ASYNC_LDS 0
#endif

__device__ __forceinline__ void stage4(float* dst_lds, const float* src) {
#if ASYNC_LDS
  __builtin_amdgcn_global_load_async_to_lds_b128((v4i*)src, (v4i*)dst_lds, 0, 0);
#else
  *(float4*)dst_lds = *(const float4*)src;
#endif
}

__device__ __forceinline__ void stage_fence() {
#if ASYNC_LDS
#if __has_builtin(__builtin_amdgcn_s_wait_asynccnt)
  __builtin_amdgcn_s_wait_asynccnt(0);
#else
  asm volatile("s_wait_asynccnt 0" ::: "memory");
#endif
#endif
}

// ---------------------------------------------------------------------------
// fc1: 1x1 conv 512->512 as GEMM (M=512, N=B*HW=16384, K=512), output written
// channel-interleaved: conv out o -> t channel (o%256)*2 + (o/256).
// Input channel k<256 -> query, else x. N-tile = 32 pixels: each A fragment
// feeds two WMMAs (halves L2 weight traffic vs 16-pixel tiles).
// ---------------------------------------------------------------------------
__global__ __launch_bounds__(256) void k_fc1(const float* __restrict__ q,
                                             const float* __restrict__ x,
                                             const float* __restrict__ w,
                                             const float* __restrict__ bias,
                                             float* __restrict__ t) {
  __shared__ float Bs[512][32];  // 64 KB input tile: K x 32 pixels
  const int n0 = blockIdx.x * 32;
  const int tid = threadIdx.x;
  // stage K x 32 tile into LDS in 16-byte chunks (async on gfx1250)
  for (int i = tid; i < 512 * 8; i += 256) {   // 8 float4 groups per K row
    int k = i >> 3, c4 = (i & 7) * 4;
    int nn = n0 + c4;
    int b = nn >> 12, pix = nn & 4095;
    const float* src = (k < 256) ? q : x;
    stage4(&Bs[k][c4], src + (b * 256 + (k & 255)) * HW + pix);
  }
  stage_fence();
  __syncthreads();
  const int wave = tid >> 5, lane = tid & 31;
  const int ml = lane & 15, hf = lane >> 4;
  for (int mt = wave; mt < 32; mt += 8) {
    const int m0 = mt * 16;
    v8f acc0 = {}, acc1 = {};
    const float* wrow = w + (m0 + ml) * 512;
    __builtin_prefetch(wrow, 0, 3);            // global_prefetch_b8
    __builtin_prefetch(wrow + 256, 0, 3);
#pragma unroll 4
    for (int kk = 0; kk < 512; kk += 4) {
      const int k = kk + hf * 2;
      v2f a, b0, b1;
      a[0] = wrow[k];
      a[1] = wrow[k + 1];
      b0[0] = Bs[k][ml];
      b0[1] = Bs[k + 1][ml];
      b1[0] = Bs[k][16 + ml];
      b1[1] = Bs[k + 1][16 + ml];
      acc0 = WMMA_F32(a, b0, acc0);
      acc1 = WMMA_F32(a, b1, acc1);
    }
    const int nnA = n0 + ml, nnB = nnA + 16;
    const int bA = nnA >> 12, pixA = nnA & 4095;
    const int bB = nnB >> 12, pixB = nnB & 4095;
#pragma unroll
    for (int r = 0; r < 8; ++r) {
      int o = m0 + r + hf * 8;
      int ch = ((o & 255) << 1) | (o >> 8);  // interleave
      float bo = bias[o];
      t[(bA * 512 + ch) * HW + pixA] = acc0[r] + bo;
      t[(bB * 512 + ch) * HW + pixB] = acc1[r] + bo;
    }
  }
}

// ---------------------------------------------------------------------------
// gate: g = sigmoid(depthwise 3x3, 2 in-ch/group over interleaved t); then
// h = query*g + x*(1-g)
// ---------------------------------------------------------------------------
__global__ __launch_bounds__(256) void k_gate(const float* __restrict__ t,
                                              const float* __restrict__ dww,
                                              const float* __restrict__ dwb,
                                              const float* __restrict__ q,
                                              const float* __restrict__ x,
                                              float* __restrict__ h) {
  int idx = blockIdx.x * 256 + threadIdx.x;  // 4*256*4096
  int pix = idx & 4095, c = (idx >> 12) & 255, b = idx >> 20;
  int py = pix >> 6, px = pix & 63;
  float acc = dwb[c];
#pragma unroll
  for (int j = 0; j < 2; ++j) {
    const float* tc = t + (b * 512 + 2 * c + j) * HW;
    const float* wj = dww + (c * 2 + j) * 9;
#pragma unroll
    for (int ky = 0; ky < 3; ++ky) {
      int yy = py + ky - 1;
      if (yy < 0 || yy > 63) continue;
#pragma unroll
      for (int kx = 0; kx < 3; ++kx) {
        int xx = px + kx - 1;
        if (xx < 0 || xx > 63) continue;
        acc += wj[ky * 3 + kx] * tc[yy * 64 + xx];
      }
    }
  }
  float g = 1.f / (1.f + __expf(-acc));
  h[idx] = q[idx] * g + x[idx] * (1.f - g);
}

// ---------------------------------------------------------------------------
// depthwise 3x3, 1 in-ch/group
// ---------------------------------------------------------------------------
__global__ __launch_bounds__(256) void k_dw(const float* __restrict__ hin,
                                            const float* __restrict__ w9,
                                            const float* __restrict__ bias,
                                            float* __restrict__ hout) {
  int idx = blockIdx.x * 256 + threadIdx.x;
  int pix = idx & 4095, c = (idx >> 12) & 255, b = idx >> 20;
  int py = pix >> 6, px = pix & 63;
  float acc = bias[c];
  const float* hc = hin + (b * 256 + c) * HW;
  const float* wj = w9 + c * 9;
#pragma unroll
  for (int ky = 0; ky < 3; ++ky) {
    int yy = py + ky - 1;
    if (yy < 0 || yy > 63) continue;
#pragma unroll
    for (int kx = 0; kx < 3; ++kx) {
      int xx = px + kx - 1;
      if (xx < 0 || xx > 63) continue;
      acc += wj[ky * 3 + kx] * hc[yy * 64 + xx];
    }
  }
  hout[idx] = acc;
}

// ---------------------------------------------------------------------------
// GroupNorm stats: one block per (b, group); 16 channels * 4096 pix each
// ---------------------------------------------------------------------------
__global__ __launch_bounds__(256) void k_gnstats(const float* __restrict__ h2,
                                                 float* __restrict__ stats) {
  __shared__ float rs[256], rq[256];
  const int bg = blockIdx.x;  // 64 = B*16
  const float* base = h2 + bg * 65536;
  float s = 0.f, q2 = 0.f;
  for (int i = threadIdx.x; i < 65536; i += 256) {
    float v = base[i];
    s += v;
    q2 += v * v;
  }
  rs[threadIdx.x] = s;
  rq[threadIdx.x] = q2;
  __syncthreads();
  for (int st = 128; st > 0; st >>= 1) {
    if (threadIdx.x < st) {
      rs[threadIdx.x] += rs[threadIdx.x + st];
      rq[threadIdx.x] += rq[threadIdx.x + st];
    }
    __syncthreads();
  }
  if (threadIdx.x == 0) {
    float m = rs[0] * (1.f / 65536.f);
    float var = rq[0] * (1.f / 65536.f) - m * m;
    stats[bg * 2] = m;
    stats[bg * 2 + 1] = rsqrtf(var + 1e-5f);
  }
}

__global__ __launch_bounds__(256) void k_gnnorm(const float* __restrict__ h2,
                                                const float* __restrict__ stats,
                                                const float* __restrict__ gw,
                                                const float* __restrict__ gb,
                                                float* __restrict__ h) {
  int idx = blockIdx.x * 256 + threadIdx.x;
  int c = (idx >> 12) & 255, b = idx >> 20;
  int g = c >> 4;
  float m = stats[(b * 16 + g) * 2];
  float r = stats[(b * 16 + g) * 2 + 1];
  float v = (h2[idx] - m) * r * gw[c] + gb[c];
  h[idx] = v / (1.f + __expf(-v));  // SiLU
}

// ---------------------------------------------------------------------------
// bot conv 256->16 + tanh + offset -> sampling grid loc[B][NP][HW][2]
// ---------------------------------------------------------------------------
__global__ __launch_bounds__(256) void k_off(const float* __restrict__ hbuf,
                                             const float* __restrict__ bw,
                                             const float* __restrict__ bb,
                                             const float* __restrict__ ref,
                                             float* __restrict__ loc) {
  int idx = blockIdx.x * 256 + threadIdx.x;  // 16384 = B*HW
  int pix = idx & 4095, b = idx >> 12;
  float acc[16];
#pragma unroll
  for (int o = 0; o < 16; ++o) acc[o] = bb[o];
  const float* hp = hbuf + b * 256 * HW + pix;
  for (int ci = 0; ci < 256; ++ci) {
    float v = hp[ci * HW];
    const float* wr = bw + ci;  // bw[o*256 + ci]
#pragma unroll
    for (int o = 0; o < 16; ++o) acc[o] += wr[o * 256] * v;
  }
  float rx = ref[pix * 2], ry = ref[pix * 2 + 1];
#pragma unroll
  for (int p = 0; p < 8; ++p) {
    float lx = rx + tanhf(acc[2 * p]) * 0.0546875f;      // SF/(2W)=7/128
    float ly = ry + tanhf(acc[2 * p + 1]) * 0.0546875f;  // SF/(2H)=7/128
    lx = fminf(1.f, fmaxf(-1.f, lx));
    ly = fminf(1.f, fmaxf(-1.f, ly));
    float* lo = loc + ((b * 8 + p) * HW + pix) * 2;
    lo[0] = lx;
    lo[1] = ly;
  }
}

// ---------------------------------------------------------------------------
// Fused deformable attention: per wave, one (b, head, 16-pixel tile).
//  1) bilinear-sample x channels [head*32, head*32+32) at 8 points -> LDS
//  2) Q = qw(32x32) @ query  via WMMA (two 16x16 C tiles)
//  3) per point: K via WMMA, logit = <Q,K> via fragment dot + shfl_xor(16)
//  4) softmax over 8 points (per lane)
//  5) out = sum_p attn[p] * (V_p via WMMA)
// Never materializes sampled/k/v tensors (saves ~380 MB of HBM/L2 traffic).
// ---------------------------------------------------------------------------
__global__ __launch_bounds__(128) void k_attn(
    const float* __restrict__ query, const float* __restrict__ xin,
    const float* __restrict__ loc, const float* __restrict__ qw,
    const float* __restrict__ qb, const float* __restrict__ kw,
    const float* __restrict__ kb, const float* __restrict__ vw,
    const float* __restrict__ vb, float* __restrict__ out) {
  __shared__ float s[4][NPt][DPH][16];  // 64 KB: per-wave sampled tiles
  const int wave = threadIdx.x >> 5, lane = threadIdx.x & 31;
  const int gt = blockIdx.x * 4 + wave;       // 8192 wave-tiles
  const int tile = gt & 255;
  const int head = (gt >> 8) & 7;
  const int b = gt >> 11;
  const int n = lane & 15;    // column (pixel) / A-matrix row index
  const int hf = lane >> 4;   // lane half
  const int pix = tile * 16 + n;
  float(*sw)[DPH][16] = s[wave];

  // ---- 1) bilinear sampling: lane handles pixel n, rows hf*16..hf*16+15
  for (int p = 0; p < NPt; ++p) {
    const float* lp = loc + ((b * NPt + p) * HW + pix) * 2;
    float xs = ((lp[0] + 1.f) * 64.f - 1.f) * 0.5f;
    float ys = ((lp[1] + 1.f) * 64.f - 1.f) * 0.5f;
    float x0f = floorf(xs), y0f = floorf(ys);
    float wx = xs - x0f, wy = ys - y0f;
    int x0 = (int)x0f, y0 = (int)y0f;
    int x1 = x0 + 1, y1 = y0 + 1;
    float vx0 = (x0 >= 0 && x0 < 64) ? 1.f : 0.f;
    float vx1 = (x1 >= 0 && x1 < 64) ? 1.f : 0.f;
    float vy0 = (y0 >= 0 && y0 < 64) ? 1.f : 0.f;
    float vy1 = (y1 >= 0 && y1 < 64) ? 1.f : 0.f;
    int cx0 = min(max(x0, 0), 63), cx1 = min(max(x1, 0), 63);
    int cy0 = min(max(y0, 0), 63), cy1 = min(max(y1, 0), 63);
    float w00 = (1.f - wx) * (1.f - wy) * vx0 * vy0;
    float w10 = wx * (1.f - wy) * vx1 * vy0;
    float w01 = (1.f - wx) * wy * vx0 * vy1;
    float w11 = wx * wy * vx1 * vy1;
    const float* img0 = xin + (b * Cn + head * DPH + hf * 16) * HW;
#pragma unroll
    for (int dd = 0; dd < 16; ++dd) {
      const float* img = img0 + dd * HW;
      float v = img[cy0 * 64 + cx0] * w00 + img[cy0 * 64 + cx1] * w10 +
                img[cy1 * 64 + cx0] * w01 + img[cy1 * 64 + cx1] * w11;
      sw[p][hf * 16 + dd][n] = v;
    }
  }
  __syncthreads();

  // ---- 2) Q projection: 32x16 output = qw(32x32) @ query-tile(32x16)
  v8f qa[2] = {{}, {}};
#pragma unroll
  for (int kk = 0; kk < 32; kk += 4) {
    const int k = kk + hf * 2;
    v2f bf;
    bf[0] = query[(b * Cn + head * DPH + k) * HW + pix];
    bf[1] = query[(b * Cn + head * DPH + k + 1) * HW + pix];
#pragma unroll
    for (int mt = 0; mt < 2; ++mt) {
      v2f a;
      const float* wr = qw + (head * DPH + mt * 16 + n) * DPH;
      a[0] = wr[k];
      a[1] = wr[k + 1];
      qa[mt] = WMMA_F32(a, bf, qa[mt]);
    }
  }
#pragma unroll
  for (int mt = 0; mt < 2; ++mt)
#pragma unroll
    for (int r = 0; r < 8; ++r)
      qa[mt][r] += qb[head * DPH + mt * 16 + r + hf * 8];

  // ---- 3) logits
  float lg[NPt];
  for (int p = 0; p < NPt; ++p) {
    v8f ka[2] = {{}, {}};
#pragma unroll
    for (int kk = 0; kk < 32; kk += 4) {
      const int k = kk + hf * 2;
      v2f bf;
      bf[0] = sw[p][k][n];
      bf[1] = sw[p][k + 1][n];
#pragma unroll
      for (int mt = 0; mt < 2; ++mt) {
        v2f a;
        const float* wr = kw + ((p * 8 + head) * DPH + mt * 16 + n) * DPH;
        a[0] = wr[k];
        a[1] = wr[k + 1];
        ka[mt] = WMMA_F32(a, bf, ka[mt]);
      }
    }
    float acc = 0.f;
#pragma unroll
    for (int mt = 0; mt < 2; ++mt)
#pragma unroll
      for (int r = 0; r < 8; ++r)
        acc += qa[mt][r] * (ka[mt][r] + kb[(p * 8 + head) * DPH + mt * 16 + r + hf * 8]);
    acc += __shfl_xor(acc, 16, 32);        // combine row halves -> full dot
    lg[p] = acc * 0.1767766952966369f;     // 1/sqrt(32)
  }

  // ---- 4) softmax over 8 points (identical on both lane halves)
  float mx = lg[0];
#pragma unroll
  for (int p = 1; p < NPt; ++p) mx = fmaxf(mx, lg[p]);
  float wgt[NPt];
  float den = 0.f;
#pragma unroll
  for (int p = 0; p < NPt; ++p) {
    wgt[p] = __expf(lg[p] - mx);
    den += wgt[p];
  }
  float inv = 1.f / den;
#pragma unroll
  for (int p = 0; p < NPt; ++p) wgt[p] *= inv;

  // ---- 5) weighted V accumulation
  v8f oa[2] = {{}, {}};
  for (int p = 0; p < NPt; ++p) {
    v8f va[2] = {{}, {}};
#pragma unroll
    for (int kk = 0; kk < 32; kk += 4) {
      const int k = kk + hf * 2;
      v2f bf;
      bf[0] = sw[p][k][n];
      bf[1] = sw[p][k + 1][n];
#pragma unroll
      for (int mt = 0; mt < 2; ++mt) {
        v2f a;
        const float* wr = vw + ((p * 8 + head) * DPH + mt * 16 + n) * DPH;
        a[0] = wr[k];
        a[1] = wr[k + 1];
        va[mt] = WMMA_F32(a, bf, va[mt]);
      }
    }
    float wp = wgt[p];
#pragma unroll
    for (int mt = 0; mt < 2; ++mt)
#pragma unroll
      for (int r = 0; r < 8; ++r)
        oa[mt][r] += wp * (va[mt][r] + vb[(p * 8 + head) * DPH + mt * 16 + r + hf * 8]);
  }
#pragma unroll
  for (int mt = 0; mt < 2; ++mt)
#pragma unroll
    for (int r = 0; r < 8; ++r)
      out[(b * Cn + head * DPH + mt * 16 + r + hf * 8) * HW + pix] = oa[mt][r];
}

// ---------------------------------------------------------------------------
// o-proj: 1x1 conv 256->256 as GEMM (M=256, N=16384, K=256), N-tile = 32
// ---------------------------------------------------------------------------
__global__ __launch_bounds__(256) void k_oproj(const float* __restrict__ in,
                                               const float* __restrict__ w,
                                               const float* __restrict__ bias,
                                               float* __restrict__ out) {
  __shared__ float Bs[256][32];  // 32 KB
  const int n0 = blockIdx.x * 32;
  const int tid = threadIdx.x;
  for (int i = tid; i < 256 * 8; i += 256) {
    int k = i >> 3, c4 = (i & 7) * 4;
    int nn = n0 + c4;
    int b = nn >> 12, pix = nn & 4095;
    stage4(&Bs[k][c4], in + (b * 256 + k) * HW + pix);
  }
  stage_fence();
  __syncthreads();
  const int wave = tid >> 5, lane = tid & 31;
  const int ml = lane & 15, hf = lane >> 4;
  for (int mt = wave; mt < 16; mt += 8) {
    const int m0 = mt * 16;
    v8f acc0 = {}, acc1 = {};
    const float* wrow = w + (m0 + ml) * 256;
    __builtin_prefetch(wrow, 0, 3);
#pragma unroll 4
    for (int kk = 0; kk < 256; kk += 4) {
      const int k = kk + hf * 2;
      v2f a, b0, b1;
      a[0] = wrow[k];
      a[1] = wrow[k + 1];
      b0[0] = Bs[k][ml];
      b0[1] = Bs[k + 1][ml];
      b1[0] = Bs[k][16 + ml];
      b1[1] = Bs[k + 1][16 + ml];
      acc0 = WMMA_F32(a, b0, acc0);
      acc1 = WMMA_F32(a, b1, acc1);
    }
    const int nnA = n0 + ml, nnB = nnA + 16;
    const int bA = nnA >> 12, pixA = nnA & 4095;
    const int bB = nnB >> 12, pixB = nnB & 4095;
#pragma unroll
    for (int r = 0; r < 8; ++r) {
      int o = m0 + r + hf * 8;
      float bo = bias[o];
      out[(bA * 256 + o) * HW + pixA] = acc0[r] + bo;
      out[(bB * 256 + o) * HW + pixB] = acc1[r] + bo;
    }
  }
}

// ---------------------------------------------------------------------------
extern "C" void kernel_launch(void* const* d_in, const int* in_sizes, int n_in,
                              void* d_out, int out_size, void* d_ws,
                              size_t ws_size, hipStream_t stream) {
  const float* query = (const float*)d_in[0];
  const float* x     = (const float*)d_in[1];
  const float* ref   = (const float*)d_in[2];
  const float* fc1w  = (const float*)d_in[3];
  const float* fc1b  = (const float*)d_in[4];
  const float* dww   = (const float*)d_in[5];
  const float* dwb   = (const float*)d_in[6];
  const float* dwbw  = (const float*)d_in[7];
  const float* dwbb  = (const float*)d_in[8];
  const float* gnw   = (const float*)d_in[9];
  const float* gnb   = (const float*)d_in[10];
  const float* botw  = (const float*)d_in[11];
  const float* botb  = (const float*)d_in[12];
  const float* qw    = (const float*)d_in[13];
  const float* qb    = (const float*)d_in[14];
  const float* kw    = (const float*)d_in[15];
  const float* kb    = (const float*)d_in[16];
  const float* vw    = (const float*)d_in[17];
  const float* vb    = (const float*)d_in[18];
  const float* ow    = (const float*)d_in[19];
  const float* ob    = (const float*)d_in[20];
  float* out = (float*)d_out;

  // workspace carve-up (floats): ~85 MB total
  float* t     = (float*)d_ws;      // 4*512*4096  = 8388608
  float* h     = t + 8388608;       // 4*256*4096  = 4194304
  float* h2    = h + 4194304;       // 4194304
  float* loc   = h2 + 4194304;      // 4*8*4096*2  = 262144
  float* stats = loc + 262144;      // 128
  float* aout  = stats + 128;       // 4194304

  k_fc1<<<512, 256, 0, stream>>>(query, x, fc1w, fc1b, t);
  k_gate<<<16384, 256, 0, stream>>>(t, dww, dwb, query, x, h);
  for (int it = 0; it < 2; ++it) {
    k_dw<<<16384, 256, 0, stream>>>(h, dwbw, dwbb, h2);
    k_gnstats<<<64, 256, 0, stream>>>(h2, stats);
    k_gnnorm<<<16384, 256, 0, stream>>>(h2, stats, gnw, gnb, h);
  }
  k_off<<<64, 256, 0, stream>>>(h, botw, botb, ref, loc);
  k_attn<<<2048, 128, 0, stream>>>(query, x, loc, qw, qb, kw, kb, vw, vb, aout);
  k_oproj<<<512, 256, 0, stream>>>(aout, ow, ob, out);
}